// YOLObase_67783173865496
// MI455X (gfx1250) — compile-verified
//
#include <hip/hip_runtime.h>
#include <cstdint>

#define NUM_CLASS 80
#define S_DIM 76
#define ANCH_PER_IMG (S_DIM * S_DIM * 3)              // 17328
#define FPA 85                                        // floats per anchor
#define IMG_FLOATS (ANCH_PER_IMG * FPA)               // 1,472,880
#define MAXGT 150
#define TILE 256
#define TILE_FLOATS (TILE * FPA)                      // 21760
#define TILE_CHUNKS (TILE_FLOATS / 4)                 // 5440 b128 chunks
#define FULL_ITERS (TILE_CHUNKS / TILE)               // 21
#define LEFTOVER (TILE_CHUNKS - FULL_ITERS * TILE)    // 64
#define BLOCKS_PER_IMG ((ANCH_PER_IMG + TILE - 1) / TILE)  // 68
#define INV_INPUT_SZ2 (1.0f / 369664.0f)              // 1/608^2

// ---- LDS float-index layout (dynamic LDS only => base offset 0) ----
#define L_CONV  0
#define L_LABEL (L_CONV + TILE_FLOATS)                // 21760
#define L_BRAW  (L_LABEL + TILE_FLOATS)               // 43520
#define L_BL    (L_BRAW + MAXGT * 4)                  // 44120
#define L_BT    (L_BL + MAXGT)
#define L_BR    (L_BT + MAXGT)
#define L_BB    (L_BR + MAXGT)
#define L_BA    (L_BB + MAXGT)
#define L_RED   (L_BA + MAXGT)                        // 44870
#define LDS_FLOATS (L_RED + 32)
#define LDS_BYTES  (LDS_FLOATS * 4)                   // ~179.6 KB < 320 KB

typedef __attribute__((ext_vector_type(2))) float v2f;
typedef __attribute__((ext_vector_type(8))) float v8f;

// ---------------- CDNA5 async global->LDS staging ----------------
__device__ __forceinline__ void async_copy_b128(uint32_t lds_byte_off,
                                                const void* gbase,
                                                uint32_t gbyte_off) {
    // GLOBAL_LOAD_ASYNC_TO_LDS_B128, saddr form: vdst(LDS addr), vaddr(offset), saddr
    asm volatile("global_load_async_to_lds_b128 %0, %1, %2"
                 :: "v"(lds_byte_off), "v"(gbyte_off), "s"(gbase)
                 : "memory");
}

__device__ __forceinline__ void wait_async_all() {
    asm volatile("s_wait_asynccnt 0" ::: "memory");
}

// ---------------- WMMA-based wave32 sum reduction ----------------
// A(16x4) = {p, 0} per lane; B(4x16) = ones; D row m = p[m] + p[m+16].
// Per-lane sum of the 8 D VGPRs covers rows 0..7 (lanes 0-15) / 8..15 (lanes 16-31);
// one shfl_xor(16) completes the full 32-lane sum in every lane.
__device__ __forceinline__ float wave_sum_wmma(float p) {
    v2f a; a[0] = p;   a[1] = 0.0f;
    v2f b; b[0] = 1.0f; b[1] = 1.0f;
    v8f c = {0.f, 0.f, 0.f, 0.f, 0.f, 0.f, 0.f, 0.f};
    v8f d = __builtin_amdgcn_wmma_f32_16x16x4_f32(false, a, false, b,
                                                  (short)0, c, false, false);
    float s = d[0] + d[1] + d[2] + d[3] + d[4] + d[5] + d[6] + d[7];
    s += __shfl_xor(s, 16, 32);
    return s;
}

__device__ __forceinline__ float sigmoidf(float x) {
    return 1.0f / (1.0f + __expf(-x));
}
__device__ __forceinline__ float bce_logits(float z, float l) {
    // max(z,0) - z*l + log1p(exp(-|z|))
    return fmaxf(z, 0.0f) - z * l + log1pf(__expf(-fabsf(z)));
}

__global__ __launch_bounds__(TILE)
void yolo_loss_main(const float* __restrict__ conv,
                    const float* __restrict__ label,
                    const float* __restrict__ bboxes,
                    float* __restrict__ part, int batch) {
    extern __shared__ float smem[];
    const int t  = threadIdx.x;
    const int bx = blockIdx.x;   // tile within image
    const int b  = blockIdx.y;   // image

    // ---- async-stage conv & label tiles (contiguous 85 KB spans) ----
    const uint32_t tileF  = (uint32_t)b * (uint32_t)IMG_FLOATS
                          + (uint32_t)bx * (uint32_t)TILE_FLOATS;
    const uint32_t maxOff = ((uint32_t)batch * (uint32_t)IMG_FLOATS - 4u) * 4u;

    for (int j = 0; j < FULL_ITERS; ++j) {
        uint32_t idx  = (uint32_t)t + (uint32_t)j * (uint32_t)TILE;
        uint32_t goff = (tileF + idx * 4u) * 4u;
        if (goff > maxOff) goff = maxOff;   // clamp tail-tile overrun (masked later)
        async_copy_b128((uint32_t)(L_CONV * 4) + idx * 16u, conv, goff);
        async_copy_b128((uint32_t)(L_LABEL * 4) + idx * 16u, label, goff);
    }
    if (t < LEFTOVER) {
        uint32_t idx  = (uint32_t)(FULL_ITERS * TILE) + (uint32_t)t;
        uint32_t goff = (tileF + idx * 4u) * 4u;
        if (goff > maxOff) goff = maxOff;
        async_copy_b128((uint32_t)(L_CONV * 4) + idx * 16u, conv, goff);
        async_copy_b128((uint32_t)(L_LABEL * 4) + idx * 16u, label, goff);
    }
    if (t < MAXGT) {   // this image's GT boxes: 150 x float4
        uint32_t goff = ((uint32_t)b * (uint32_t)(MAXGT * 4) + (uint32_t)t * 4u) * 4u;
        async_copy_b128((uint32_t)(L_BRAW * 4) + (uint32_t)t * 16u, bboxes, goff);
    }
    wait_async_all();
    __syncthreads();

    // ---- GT boxes -> corner + area form (read as broadcasts later) ----
    if (t < MAXGT) {
        float x = smem[L_BRAW + t * 4 + 0];
        float y = smem[L_BRAW + t * 4 + 1];
        float w = smem[L_BRAW + t * 4 + 2];
        float h = smem[L_BRAW + t * 4 + 3];
        smem[L_BL + t] = x - 0.5f * w;
        smem[L_BT + t] = y - 0.5f * h;
        smem[L_BR + t] = x + 0.5f * w;
        smem[L_BB + t] = y + 0.5f * h;
        smem[L_BA + t] = w * h;
    }
    __syncthreads();

    // ---- per-anchor compute ----
    const int ai = bx * TILE + t;
    const bool valid = (ai < ANCH_PER_IMG);
    const float* cv = &smem[L_CONV + t * FPA];
    const float* lb = &smem[L_LABEL + t * FPA];

    const int a    = ai % 3;
    const int cell = ai / 3;
    const int gx   = cell % S_DIM;
    const int gy   = cell / S_DIM;

    const float AW[3] = {12.f, 19.f, 40.f};
    const float AH[3] = {16.f, 36.f, 28.f};

    // decode
    float px = (sigmoidf(cv[0]) * 1.2f - 0.1f + (float)gx) * 8.0f;
    float py = (sigmoidf(cv[1]) * 1.2f - 0.1f + (float)gy) * 8.0f;
    float pw = __expf(cv[2]) * AW[a];
    float ph = __expf(cv[3]) * AH[a];
    float parea = pw * ph;
    float pl = px - 0.5f * pw, pr = px + 0.5f * pw;
    float pt = py - 0.5f * ph, pb = py + 0.5f * ph;

    // label box
    float lx = lb[0], ly = lb[1], lw = lb[2], lh = lb[3], resp = lb[4];
    float larea = lw * lh;
    float ll = lx - 0.5f * lw, lr = lx + 0.5f * lw;
    float lt = ly - 0.5f * lh, lbm = ly + 0.5f * lh;

    // GIoU
    float ix = fmaxf(fminf(pr, lr) - fmaxf(pl, ll), 0.0f);
    float iy = fmaxf(fminf(pb, lbm) - fmaxf(pt, lt), 0.0f);
    float inter = ix * iy;
    float uni   = parea + larea - inter;
    float iou   = inter / uni;
    float ex = fmaxf(fmaxf(pr, lr) - fminf(pl, ll), 0.0f);
    float ey = fmaxf(fmaxf(pb, lbm) - fminf(pt, lt), 0.0f);
    float enca = ex * ey;
    float giou = iou - (enca - uni) / enca;
    float pg = resp * (2.0f - larea * INV_INPUT_SZ2) * (1.0f - giou);

    // max-IoU-vs-GT predicate, divide-free: exists t with 2*inter >= union
    float mhit = -3.402823466e38f;
    for (int k = 0; k < MAXGT; ++k) {
        float qx = fmaxf(fminf(pr, smem[L_BR + k]) - fmaxf(pl, smem[L_BL + k]), 0.0f);
        float qy = fmaxf(fminf(pb, smem[L_BB + k]) - fmaxf(pt, smem[L_BT + k]), 0.0f);
        float qi = qx * qy;
        float qu = parea + smem[L_BA + k] - qi;
        mhit = fmaxf(mhit, 2.0f * qi - qu);   // >=0 <=> iou >= 0.5
    }
    float bgd = (mhit >= 0.0f) ? 0.0f : (1.0f - resp);

    // conf loss
    float zc = cv[4];
    float df = resp - sigmoidf(zc);
    float pc = (df * df) * (resp + bgd) * bce_logits(zc, resp);

    // prob loss
    float pp = 0.0f;
    for (int c2 = 0; c2 < NUM_CLASS; ++c2)
        pp += bce_logits(cv[5 + c2], lb[5 + c2]);
    pp *= resp;

    // select (not multiply) so garbage NaNs in tail anchors cannot leak
    pg = valid ? pg : 0.0f;
    pc = valid ? pc : 0.0f;
    pp = valid ? pp : 0.0f;

    // ---- WMMA wave reduction (EXEC all-1s: uniform path, no early exits) ----
    float sg = wave_sum_wmma(pg);
    float sc = wave_sum_wmma(pc);
    float sp = wave_sum_wmma(pp);

    const int wave = t >> 5, lane = t & 31;
    if (lane == 0) {
        smem[L_RED + wave * 3 + 0] = sg;
        smem[L_RED + wave * 3 + 1] = sc;
        smem[L_RED + wave * 3 + 2] = sp;
    }
    __syncthreads();
    if (t < 3) {
        float s = 0.0f;
        for (int w = 0; w < 8; ++w) s += smem[L_RED + w * 3 + t];
        part[(b * gridDim.x + bx) * 3 + t] = s;
    }
}

__global__ __launch_bounds__(256)
void yolo_loss_reduce(const float* __restrict__ part, int nblocks, float invB,
                      float* __restrict__ out) {
    __shared__ float red[24];
    float s0 = 0.f, s1 = 0.f, s2 = 0.f;
    for (int i = threadIdx.x; i < nblocks; i += 256) {
        s0 += part[3 * i + 0];
        s1 += part[3 * i + 1];
        s2 += part[3 * i + 2];
    }
    for (int o = 16; o > 0; o >>= 1) {
        s0 += __shfl_xor(s0, o, 32);
        s1 += __shfl_xor(s1, o, 32);
        s2 += __shfl_xor(s2, o, 32);
    }
    const int wave = threadIdx.x >> 5, lane = threadIdx.x & 31;
    if (lane == 0) {
        red[wave * 3 + 0] = s0;
        red[wave * 3 + 1] = s1;
        red[wave * 3 + 2] = s2;
    }
    __syncthreads();
    if (threadIdx.x < 3) {
        float s = 0.0f;
        for (int w = 0; w < 8; ++w) s += red[w * 3 + threadIdx.x];
        out[threadIdx.x] = s * invB;
    }
}

extern "C" void kernel_launch(void* const* d_in, const int* in_sizes, int n_in,
                              void* d_out, int out_size, void* d_ws, size_t ws_size,
                              hipStream_t stream) {
    const float* conv   = (const float*)d_in[0];
    const float* label  = (const float*)d_in[1];
    const float* bboxes = (const float*)d_in[2];
    float* out  = (float*)d_out;
    float* part = (float*)d_ws;

    int batch = in_sizes[0] / IMG_FLOATS;
    if (batch < 1) batch = 1;

    (void)hipFuncSetAttribute(reinterpret_cast<const void*>(yolo_loss_main),
                              hipFuncAttributeMaxDynamicSharedMemorySize, LDS_BYTES);

    dim3 grid(BLOCKS_PER_IMG, batch);
    yolo_loss_main<<<grid, TILE, LDS_BYTES, stream>>>(conv, label, bboxes, part, batch);

    const int nblocks = BLOCKS_PER_IMG * batch;
    yolo_loss_reduce<<<1, 256, 0, stream>>>(part, nblocks, 1.0f / (float)batch, out);
}